// ToDenseBEVConvolution_14594298871921
// MI455X (gfx1250) — compile-verified
//
#include <hip/hip_runtime.h>

typedef float v2f __attribute__((ext_vector_type(2)));
typedef float v8f __attribute__((ext_vector_type(8)));

#define NKERN 32
#define C_IN  64
#define C_OUT 64
#define BEVX  256
#define BEVY  256

// Workspace layout (as int*):
//   [0..31]    counts
//   [32..64]   bucketStart (33, exclusive prefix; [64] = N)
//   [65..97]   tileStart   (33; [97] = total tiles)
//   [98..129]  cursor      (init = bucketStart)
//   [130..]    sortedIdx   (N ints)
#define WS_COUNTS 0
#define WS_BSTART 32
#define WS_TSTART 65
#define WS_CURSOR 98
#define WS_SORTED 130

__global__ void hist_kernel(const int* __restrict__ coords, const int* __restrict__ strideP,
                            int n, int* __restrict__ ws) {
    int i = blockIdx.x * blockDim.x + threadIdx.x;
    if (i < n) {
        int z = coords[4 * i + 1] / strideP[0];
        atomicAdd(&ws[WS_COUNTS + z], 1);
    }
}

__global__ void prefix_kernel(int* __restrict__ ws) {
    if (threadIdx.x == 0) {
        int acc = 0, tacc = 0;
        for (int z = 0; z < NKERN; ++z) {
            ws[WS_BSTART + z] = acc;
            ws[WS_TSTART + z] = tacc;
            ws[WS_CURSOR + z] = acc;
            int c = ws[WS_COUNTS + z];
            acc  += c;
            tacc += (c + 15) >> 4;
        }
        ws[WS_BSTART + NKERN] = acc;
        ws[WS_TSTART + NKERN] = tacc;
    }
}

__global__ void scatter_kernel(const int* __restrict__ coords, const int* __restrict__ strideP,
                               int n, int* __restrict__ ws) {
    int i = blockIdx.x * blockDim.x + threadIdx.x;
    if (i < n) {
        int z = coords[4 * i + 1] / strideP[0];
        int pos = atomicAdd(&ws[WS_CURSOR + z], 1);  // absolute position (cursor starts at bucketStart)
        ws[WS_SORTED + pos] = i;
    }
}

// One wave (32 threads) per tile of up to 16 points sharing the same kernel z.
// Computes D(16x64) = A(16x64 features) x B(64x64 kernel) via 64x v_wmma_f32_16x16x4_f32,
// then scatter-adds into out laid out as (B, C_OUT, BEVX, BEVY).
__global__ void __launch_bounds__(32) gemm_scatter_kernel(
    const float* __restrict__ feats, const float* __restrict__ kern,
    const int* __restrict__ coords, const int* __restrict__ strideP,
    const int* __restrict__ ws, float* __restrict__ out)
{
    const int tileId = blockIdx.x;
    const int totalTiles = ws[WS_TSTART + NKERN];
    if (tileId >= totalTiles) return;  // uniform per wave: EXEC stays all-1s for WMMA

    // Find bucket z for this tile (32-entry scan, uniform/scalar).
    int z = 0;
    for (int zz = 1; zz < NKERN; ++zz)
        if (ws[WS_TSTART + zz] <= tileId) z = zz;
    const int localTile = tileId - ws[WS_TSTART + z];
    const int bStart = ws[WS_BSTART + z];
    const int bEnd   = ws[WS_BSTART + z + 1];
    const int pBase  = bStart + localTile * 16;
    const int nPts   = min(16, bEnd - pBase);

    const int lane = threadIdx.x;
    const int half = lane >> 4;       // 0: K lanes {0,1}; 1: K lanes {2,3}
    const int l15  = lane & 15;
    const int koff = half * 2;
    const int stride = strideP[0];
    const int* __restrict__ sortedIdx = ws + WS_SORTED;

    // ---- Load A (16x64 f32): lane holds row (l15), K pair per k-step ----
    const int arow = l15;
    const int ap = sortedIdx[pBase + min(arow, nPts - 1)];
    const float* __restrict__ fRow = feats + (size_t)ap * C_IN;
    v2f a[16];
#pragma unroll
    for (int kk = 0; kk < 16; ++kk) {
        const float2 t = *(const float2*)(fRow + 4 * kk + koff);
        a[kk].x = t.x; a[kk].y = t.y;
    }

    // ---- Precompute per-row output bases: out[((b*64+c)*256+x)*256+y] ----
    int base[8];
#pragma unroll
    for (int i = 0; i < 8; ++i) {
        const int m = i + 8 * half;            // C/D row for VGPR i on this lane half
        const bool valid = (m < nPts);
        const int p = sortedIdx[pBase + (valid ? m : 0)];
        const int x = coords[4 * p + 0] / stride;
        const int y = coords[4 * p + 2] / stride;
        const int b = coords[4 * p + 3];
        base[i] = valid ? (b * (C_OUT * BEVX * BEVY) + x * BEVY + y) : -1;
    }

    const float* __restrict__ K = kern + (size_t)z * C_IN * C_OUT;

#pragma unroll
    for (int ct = 0; ct < 4; ++ct) {           // 4 column tiles of 16 channels
        const int col = ct * 16 + l15;
        v8f acc = {0.f, 0.f, 0.f, 0.f, 0.f, 0.f, 0.f, 0.f};
#pragma unroll
        for (int kk = 0; kk < 16; ++kk) {      // 16 k-steps of 4
            v2f bv;
            bv.x = K[(4 * kk + koff) * C_OUT + col];
            bv.y = K[(4 * kk + koff + 1) * C_OUT + col];
            acc = __builtin_amdgcn_wmma_f32_16x16x4_f32(
                /*neg_a=*/false, a[kk], /*neg_b=*/false, bv,
                /*c_mod=*/(short)0, acc, /*reuse_a=*/false, /*reuse_b=*/false);
        }
        // D layout: VGPR i, lanes 0-15 -> row i, lanes 16-31 -> row i+8; col = lane&15 (+16*ct)
#pragma unroll
        for (int i = 0; i < 8; ++i) {
            if (base[i] >= 0) {
                unsafeAtomicAdd(&out[(size_t)base[i] + (size_t)col * (BEVX * BEVY)], acc[i]);
            }
        }
    }
}

extern "C" void kernel_launch(void* const* d_in, const int* in_sizes, int n_in,
                              void* d_out, int out_size, void* d_ws, size_t ws_size,
                              hipStream_t stream) {
    const float* feats   = (const float*)d_in[0];
    const float* kern    = (const float*)d_in[1];
    const int*   coords  = (const int*)d_in[2];
    const int*   strideP = (const int*)d_in[3];
    // d_in[4] = batch_size (implicit in out_size)

    const int n = in_sizes[0] / C_IN;
    float* out = (float*)d_out;
    int* ws = (int*)d_ws;

    // Zero output and bucket counters (capture-safe async memsets).
    hipMemsetAsync(d_out, 0, (size_t)out_size * sizeof(float), stream);
    hipMemsetAsync(d_ws, 0, NKERN * sizeof(int), stream);

    const int tb = 256;
    const int nb = (n + tb - 1) / tb;
    hist_kernel   <<<nb, tb, 0, stream>>>(coords, strideP, n, ws);
    prefix_kernel <<<1, 32, 0, stream>>>(ws);
    scatter_kernel<<<nb, tb, 0, stream>>>(coords, strideP, n, ws);

    // Upper bound on tiles: each bucket adds at most 15 padding slots.
    const int maxTiles = (n + 15) / 16 + NKERN;
    gemm_scatter_kernel<<<maxTiles, 32, 0, stream>>>(feats, kern, coords, strideP, ws, out);
}